// GCN_1357209666152
// MI455X (gfx1250) — compile-verified
//
#include <hip/hip_runtime.h>

#define D 128

typedef __attribute__((ext_vector_type(2))) float v2f;
typedef __attribute__((ext_vector_type(8))) float v8f;

// ---------------------------------------------------------------------------
// Pack W (D x D, row-major) into per-lane WMMA-B operand order so the GEMM
// reads B as one coalesced b64 per (k-step, n-tile):
//   P[(k4*8 + nt)*32 + lane] = { W[k4*4+khalf][nt*16+row],
//                                W[k4*4+khalf+1][nt*16+row] }
// with row = lane&15, khalf = (lane>>4)*2  (ISA 7.12.2 fp32 B layout).
// ---------------------------------------------------------------------------
__global__ __launch_bounds__(256) void pack_w_kernel(
    const float* __restrict__ W, float2* __restrict__ P) {
  int i = blockIdx.x * blockDim.x + threadIdx.x;   // 0 .. 8191
  int lane = i & 31;
  int nt   = (i >> 5) & 7;
  int k4   = i >> 8;
  int row   = lane & 15;
  int khalf = (lane >> 4) << 1;
  int kr    = k4 * 4 + khalf;
  float2 p;
  p.x = W[(size_t)kr * D + nt * 16 + row];
  p.y = W[(size_t)(kr + 1) * D + nt * 16 + row];
  P[i] = p;
}

// ---------------------------------------------------------------------------
// GEMM: H = act(X) @ W  (RELU as template param -> no per-element selects).
// Packed W (64KB) is staged once per workgroup into LDS with CDNA5 async
// global->LDS copies (ASYNCcnt), then each wave computes a full 16x128 output
// panel: 8 C-tiles, A loaded once per K-step (global b64) and reused by 8
// WMMAs; B read from LDS as ds_load_b64 with immediate offsets.
// ---------------------------------------------------------------------------
template <int RELU>
__global__ __launch_bounds__(256) void gemm_wmma_f32(
    const float* __restrict__ X, const float2* __restrict__ P,
    float* __restrict__ H, int npanels) {
  __shared__ float2 shP[(D / 4) * 8 * 32];         // 8192 float2 = 64KB

  const int lane = threadIdx.x & 31;
  const int wave = threadIdx.x >> 5;

  // ---- async-stage packed W into LDS: 4096 x 16B chunks, 16 per thread ----
  {
    const unsigned long long pbase = (unsigned long long)P;
    unsigned lds0 = (unsigned)(size_t)(&shP[0]);   // low 32 bits = LDS offset
#pragma unroll
    for (int j = 0; j < 16; ++j) {
      unsigned c    = threadIdx.x + j * 256;       // chunk index
      unsigned voff = c * 16;                      // global byte offset
      unsigned ldsa = lds0 + c * 16;               // LDS byte address
      asm volatile("global_load_async_to_lds_b128 %0, %1, %2"
                   :
                   : "v"(ldsa), "v"(voff), "s"(pbase)
                   : "memory");
    }
    asm volatile("s_wait_asynccnt 0x0" ::: "memory");
  }
  __syncthreads();

  int mp = blockIdx.x * 8 + wave;
  mp = mp < npanels ? mp : npanels - 1;            // clamp: EXEC stays all-1s
  const int m0    = mp * 16;
  const int row   = lane & 15;
  const int khalf = (lane >> 4) << 1;              // 0 or 2

  const float*  xrow = X + (size_t)(m0 + row) * D + khalf;
  const float2* lp   = shP + lane;                 // + (k4*8+nt)*32 (imm offs)

  v8f acc[8] = {};
#pragma unroll 4
  for (int k4 = 0; k4 < D / 4; ++k4) {
    v2f a = *(const v2f*)(xrow + k4 * 4);          // global_load_b64
    if (RELU) {
      a.x = fmaxf(a.x, 0.f);
      a.y = fmaxf(a.y, 0.f);
    }
#pragma unroll
    for (int nt = 0; nt < 8; ++nt) {
      v2f b = *(const v2f*)(lp + (k4 * 8 + nt) * 32);   // ds_load_b64
      acc[nt] = __builtin_amdgcn_wmma_f32_16x16x4_f32(
          false, a, false, b, (short)0, acc[nt], false, false);
    }
  }

  const int rbase = (lane >> 4) << 3;              // 0 or 8
  float* out = H + (size_t)(m0 + rbase) * D + row;
#pragma unroll
  for (int nt = 0; nt < 8; ++nt)
#pragma unroll
    for (int r = 0; r < 8; ++r)
      out[(size_t)r * D + nt * 16] = acc[nt][r];
}

// ---------------------------------------------------------------------------
// Edge aggregation: one wave per edge; lane l owns features 4l..4l+3.
// agg[dst] += h[src] * dis[src]*dis[dst]  (native f32 L2 atomics).
// agg must be pre-initialized with the layer bias.
// ---------------------------------------------------------------------------
__global__ __launch_bounds__(256) void scatter_edges(
    const float* __restrict__ H, const int* __restrict__ src,
    const int* __restrict__ dst, const float* __restrict__ dis,
    float* __restrict__ agg, int E) {
  const int lane = threadIdx.x & 31;
  int e = blockIdx.x * 8 + (threadIdx.x >> 5);
  if (e >= E) return;                              // uniform per wave
  e = __builtin_amdgcn_readfirstlane(e);
  const int s = __builtin_amdgcn_readfirstlane(src[e]);
  const int d = __builtin_amdgcn_readfirstlane(dst[e]);
  const float nrm = dis[s] * dis[d];
  const float4 v  = ((const float4*)(H + (size_t)s * D))[lane];
  float* ap = agg + (size_t)d * D + lane * 4;
  __hip_atomic_fetch_add(ap + 0, v.x * nrm, __ATOMIC_RELAXED, __HIP_MEMORY_SCOPE_AGENT);
  __hip_atomic_fetch_add(ap + 1, v.y * nrm, __ATOMIC_RELAXED, __HIP_MEMORY_SCOPE_AGENT);
  __hip_atomic_fetch_add(ap + 2, v.z * nrm, __ATOMIC_RELAXED, __HIP_MEMORY_SCOPE_AGENT);
  __hip_atomic_fetch_add(ap + 3, v.w * nrm, __ATOMIC_RELAXED, __HIP_MEMORY_SCOPE_AGENT);
}

__global__ void degree_kernel(const int* __restrict__ src,
                              float* __restrict__ deg, int E) {
  int i = blockIdx.x * blockDim.x + threadIdx.x;
  if (i < E)
    __hip_atomic_fetch_add(&deg[src[i]], 1.0f, __ATOMIC_RELAXED,
                           __HIP_MEMORY_SCOPE_AGENT);
}

__global__ void rsqrt_kernel(float* __restrict__ deg, int n) {
  int i = blockIdx.x * blockDim.x + threadIdx.x;
  if (i < n) {
    float d = deg[i];
    deg[i] = d > 0.f ? rsqrtf(d) : 0.f;
  }
}

__global__ void zero_kernel(float* __restrict__ p, int n) {
  int i = blockIdx.x * blockDim.x + threadIdx.x;
  if (i < n) p[i] = 0.f;
}

// p[i] = b[i % D]  (bias broadcast init of the aggregation buffer)
__global__ void bias_init_kernel(float* __restrict__ p,
                                 const float* __restrict__ b, int n) {
  int i = blockIdx.x * blockDim.x + threadIdx.x;
  if (i < n) p[i] = b[i & (D - 1)];
}

extern "C" void kernel_launch(void* const* d_in, const int* in_sizes, int n_in,
                              void* d_out, int out_size, void* d_ws,
                              size_t ws_size, hipStream_t stream) {
  const float* x    = (const float*)d_in[0];
  const int*   edge = (const int*)d_in[1];
  const float* W0   = (const float*)d_in[2];
  const float* b0   = (const float*)d_in[3];
  const float* W1   = (const float*)d_in[4];
  const float* b1   = (const float*)d_in[5];
  const float* W2   = (const float*)d_in[6];
  const float* b2   = (const float*)d_in[7];
  float*       out  = (float*)d_out;

  const int N = in_sizes[0] / D;   // 50000
  const int E = in_sizes[1] / 2;   // 800000
  // reference: src, dst = edge_index[1], edge_index[0]
  const int* dstp = edge;
  const int* srcp = edge + E;

  const size_t ND = (size_t)N * D;
  const int    NP = (D / 4) * 8 * 32;              // packed W elems (float2)
  float*  dis = (float*)d_ws;
  float*  h   = dis + (((size_t)N + 255) & ~(size_t)255);
  float*  agg = h + ND;
  float2* P0  = (float2*)(agg + ND);
  float2* P1  = P0 + NP;
  float2* P2  = P1 + NP;

  const int T = 256;
  const dim3 blk(T);
  const int npanels = N / 16;                      // 3125
  const int gE  = (E + T - 1) / T;
  const int gN  = (N + T - 1) / T;
  const int gND = (int)((ND + T - 1) / T);
  const int gSc = (E + 7) / 8;                     // 1 wave per edge
  const int gMM = (npanels + 7) / 8;               // 8 panels per block
  const int gPk = (NP + T - 1) / T;

  // one-shot prep: degree -> dis = rsqrt(deg); pack weights
  zero_kernel<<<gN, blk, 0, stream>>>(dis, N);
  degree_kernel<<<gE, blk, 0, stream>>>(srcp, dis, E);
  rsqrt_kernel<<<gN, blk, 0, stream>>>(dis, N);
  pack_w_kernel<<<gPk, blk, 0, stream>>>(W0, P0);
  pack_w_kernel<<<gPk, blk, 0, stream>>>(W1, P1);
  pack_w_kernel<<<gPk, blk, 0, stream>>>(W2, P2);

  // ---- layer 0:  h = x @ W0 ; agg = b0 + scatter(h) ----
  gemm_wmma_f32<0><<<gMM, blk, 0, stream>>>(x, P0, h, npanels);
  bias_init_kernel<<<gND, blk, 0, stream>>>(agg, b0, (int)ND);
  scatter_edges<<<gSc, blk, 0, stream>>>(h, srcp, dstp, dis, agg, E);

  // ---- layer 1:  h = relu(agg) @ W1 ; agg = b1 + scatter(h) ----
  gemm_wmma_f32<1><<<gMM, blk, 0, stream>>>(agg, P1, h, npanels);
  bias_init_kernel<<<gND, blk, 0, stream>>>(agg, b1, (int)ND);
  scatter_edges<<<gSc, blk, 0, stream>>>(h, srcp, dstp, dis, agg, E);

  // ---- layer 2:  h = relu(agg) @ W2 ; out = b2 + scatter(h) ----
  gemm_wmma_f32<1><<<gMM, blk, 0, stream>>>(agg, P2, h, npanels);
  bias_init_kernel<<<gND, blk, 0, stream>>>(out, b2, (int)ND);
  scatter_edges<<<gSc, blk, 0, stream>>>(h, srcp, dstp, dis, out, E);
}